// ClusterLoss_5102421147886
// MI455X (gfx1250) — compile-verified
//
#include <hip/hip_runtime.h>
#include <hip/hip_bf16.h>

#define P   256     // ids per batch
#define KI  16      // imgs per id
#define D   2048    // feature dim
#define MARGIN 10.0f
#define EPSV   1e-12f

typedef float v2f __attribute__((ext_vector_type(2)));
typedef float v4f __attribute__((ext_vector_type(4)));
typedef float v8f __attribute__((ext_vector_type(8)));

// ds_swizzle group-of-32 XOR offsets: (xor_mask<<10) | and_mask(0x1f)
template <int OFFS>
__device__ __forceinline__ float swz_min(float v) {
    return fminf(v, __int_as_float(
        __builtin_amdgcn_ds_swizzle(__float_as_int(v), OFFS)));
}

// ---------------------------------------------------------------------------
// Kernel 1: per-class centers, ||c||^2, intra_max. One block per class.
// Thread t owns the contiguous columns d in [8t, 8t+8) -> two b128 loads/row.
// ---------------------------------------------------------------------------
__global__ void __launch_bounds__(256)
centers_intra_kernel(const float* __restrict__ f,
                     float* __restrict__ centers,
                     float* __restrict__ cc,
                     unsigned int* __restrict__ intermin_bits,
                     float* __restrict__ out_intra /* = d_out + 1 */) {
    const int p = blockIdx.x;
    const int t = threadIdx.x;

    __shared__ float red[KI * 256];   // 16 KB: per-k partial sums
    __shared__ float ccred[256];      // 1 KB : ||c||^2 partials

    const float* fp = f + (size_t)p * KI * D + t * 8;

    // Pass 1: centers (mean over the 16 rows of this class), b128 loads
    v4f s0 = {0.f, 0.f, 0.f, 0.f};
    v4f s1 = {0.f, 0.f, 0.f, 0.f};
#pragma unroll
    for (int k = 0; k < KI; ++k) {
        const v4f a = *reinterpret_cast<const v4f*>(fp + (size_t)k * D);
        const v4f b = *reinterpret_cast<const v4f*>(fp + (size_t)k * D + 4);
        s0 += a;
        s1 += b;
    }
    const v4f c0 = s0 * (1.0f / (float)KI);
    const v4f c1 = s1 * (1.0f / (float)KI);
    float* crow = centers + (size_t)p * D + t * 8;
    *reinterpret_cast<v4f*>(crow)     = c0;
    *reinterpret_cast<v4f*>(crow + 4) = c1;

    ccred[t] = c0.x * c0.x + c0.y * c0.y + c0.z * c0.z + c0.w * c0.w +
               c1.x * c1.x + c1.y * c1.y + c1.z * c1.z + c1.w * c1.w;

    // Pass 2: per-image squared distance to the center (tile is cache-hot)
    float part[KI];
#pragma unroll
    for (int k = 0; k < KI; ++k) {
        const v4f a  = *reinterpret_cast<const v4f*>(fp + (size_t)k * D);
        const v4f b  = *reinterpret_cast<const v4f*>(fp + (size_t)k * D + 4);
        const v4f d0 = a - c0;
        const v4f d1 = b - c1;
        part[k] = d0.x * d0.x + d0.y * d0.y + d0.z * d0.z + d0.w * d0.w +
                  d1.x * d1.x + d1.y * d1.y + d1.z * d1.z + d1.w * d1.w;
    }
#pragma unroll
    for (int k = 0; k < KI; ++k) red[k * 256 + t] = part[k];
    __syncthreads();

    for (int s = 128; s > 0; s >>= 1) {
        if (t < s) {
#pragma unroll
            for (int k = 0; k < KI; ++k) red[k * 256 + t] += red[k * 256 + t + s];
            ccred[t] += ccred[t + s];
        }
        __syncthreads();
    }

    if (t == 0) {
        float m = 0.0f;  // all distances are > 0 after the eps clip
#pragma unroll
        for (int k = 0; k < KI; ++k) {
            const float d2 = red[k * 256];
            m = fmaxf(m, sqrtf(fmaxf(d2, EPSV)));
        }
        out_intra[p]     = m;
        cc[p]            = ccred[0];
        intermin_bits[p] = 0x7F800000u;  // +inf, for the atomic-min in kernel 2
    }
}

// ---------------------------------------------------------------------------
// Kernel 2: Gram = centers @ centers^T via V_WMMA_F32_16X16X4_F32,
// then pairwise distances and per-row min (atomic min on float bits).
// 4 waves / block, one 16x16 output tile per wave, 256 tiles total.
// ---------------------------------------------------------------------------
__global__ void __launch_bounds__(128)
gram_wmma_kernel(const float* __restrict__ centers,
                 const float* __restrict__ cc,
                 unsigned int* __restrict__ intermin_bits) {
    const int wave = threadIdx.x >> 5;              // 0..3
    const int lane = threadIdx.x & 31;
    const int tile = blockIdx.x * 4 + wave;         // 0..255
    const int tm   = tile >> 4;                     // tile row
    const int tn   = tile & 15;                     // tile col
    const int half = lane >> 4;                     // 0 | 1
    const int nl   = lane & 15;                     // 0..15

    // A 16x4 fragment: lane L -> A[L%16][k0 + (L>>4)*2 + {0,1}]  (ISA 7.12.2)
    // B 4x16 fragment: lane L -> B[(L>>4)*2 + v][L%16]; B = C^T so the load
    // pattern is identical with the row taken from the tn tile.
    const float* arow = centers + (size_t)(tm * 16 + nl) * D + half * 2;
    const float* brow = centers + (size_t)(tn * 16 + nl) * D + half * 2;

    v8f acc = {};
#pragma unroll 4
    for (int k0 = 0; k0 < D; k0 += 4) {
        const v2f a = *reinterpret_cast<const v2f*>(arow + k0);
        const v2f b = *reinterpret_cast<const v2f*>(brow + k0);
        acc = __builtin_amdgcn_wmma_f32_16x16x4_f32(
            /*neg_a=*/false, a, /*neg_b=*/false, b,
            /*c_mod=*/(short)0, acc, /*reuse_a=*/false, /*reuse_b=*/false);
    }

    // C/D layout: VGPR r -> row M = half*8 + r, col N = nl.
    const int   gn  = tn * 16 + nl;
    const float ccn = cc[gn];
#pragma unroll
    for (int r = 0; r < 8; ++r) {
        const int gm = tm * 16 + (half * 8 + r);
        float d2 = cc[gm] + ccn - 2.0f * acc[r];
        float dd = sqrtf(fmaxf(d2, EPSV));
        if (gm == gn) dd = __builtin_inff();        // exclude self
        // min over the 16 lanes of this half: ds_swizzle XOR masks 8/4/2/1
        // (masks < 16 never cross the 16-lane halves in wave32)
        dd = swz_min<0x201F>(dd);   // xor 8
        dd = swz_min<0x101F>(dd);   // xor 4
        dd = swz_min<0x081F>(dd);   // xor 2
        dd = swz_min<0x041F>(dd);   // xor 1
        if (nl == 0)
            atomicMin(intermin_bits + gm, __float_as_uint(dd));
    }
}

// ---------------------------------------------------------------------------
// Kernel 3: inter_min out, loss = mean(relu(intra_max - inter_min + MARGIN))
// ---------------------------------------------------------------------------
__global__ void __launch_bounds__(256)
finalize_kernel(const float* __restrict__ intra /* = d_out + 1 */,
                const unsigned int* __restrict__ intermin_bits,
                float* __restrict__ d_out) {
    __shared__ float red[256];
    const int t = threadIdx.x;

    const float inter = __uint_as_float(intermin_bits[t]);
    d_out[1 + P + t] = inter;
    red[t] = fmaxf(intra[t] - inter + MARGIN, 0.0f);
    __syncthreads();
    for (int s = 128; s > 0; s >>= 1) {
        if (t < s) red[t] += red[t + s];
        __syncthreads();
    }
    if (t == 0) d_out[0] = red[0] * (1.0f / (float)P);
}

// ---------------------------------------------------------------------------
extern "C" void kernel_launch(void* const* d_in, const int* in_sizes, int n_in,
                              void* d_out, int out_size, void* d_ws, size_t ws_size,
                              hipStream_t stream) {
    const float* features = (const float*)d_in[0];
    // d_in[1] = targets (int64) is implied by the ordered layout; unused.
    float* out = (float*)d_out;   // [0]=loss, [1..256]=intra_max, [257..512]=inter_min

    float*        centers  = (float*)d_ws;                       // 256*2048 f32 (2 MB)
    float*        cc       = centers + (size_t)P * D;            // 256 f32
    unsigned int* intermin = (unsigned int*)(cc + P);            // 256 u32

    centers_intra_kernel<<<P, 256, 0, stream>>>(features, centers, cc, intermin, out + 1);
    gram_wmma_kernel<<<P / 4, 128, 0, stream>>>(centers, cc, intermin);
    finalize_kernel<<<1, 256, 0, stream>>>(out + 1, intermin, out);
}